// BCPLoss_31653908972139
// MI455X (gfx1250) — compile-verified
//
#include <hip/hip_runtime.h>
#include <hip/hip_bf16.h>

// ---------------- problem constants ----------------
#define BATCH 16
#define CHAN  3
#define HH    512
#define WW    512
#define HW    262144          // 512*512
#define NPIX  (BATCH * HW)    // 4,194,304
#define RAD   17              // (35-1)/2
#define KSEL  26              // int(0.0001 * 512 * 512)
#define NIDX  (BATCH * KSEL)  // 416
#define FINF  3.402823466e+38f

typedef __attribute__((ext_vector_type(4))) unsigned int u32x4;
typedef __attribute__((ext_vector_type(8))) int          i32x8;
typedef __attribute__((ext_vector_type(4))) int          i32x4;
typedef __attribute__((ext_vector_type(2))) float        f32x2;
typedef __attribute__((ext_vector_type(8))) float        f32x8;

// =====================================================================
// Kernel 1: horizontal pass. One block per (b,y) row. A single TDM
// tensor_load_to_lds moves the 3x512 f32 tile (stride H*W between
// channel lines) into LDS; threads then build the channel-max row and
// the 35-tap horizontal max (of chan-max) and per-channel horizontal min.
// =====================================================================
__global__ __launch_bounds__(256) void bcp_hpass(const float* __restrict__ img,
                                                 float* __restrict__ hmax,
                                                 float* __restrict__ hmin) {
    __shared__ float smem[2048]; // [0..1535]: 3 channel rows, [1536..2047]: chan-max row
    const int bid = blockIdx.x;          // b*512 + y
    const int b   = bid >> 9;
    const int y   = bid & 511;
    const int tid = threadIdx.x;

    // ---- TDM: issue from wave 0 only (EXEC is ignored by TDM) ----
    if (tid < 32) {
        unsigned long long gaddr =
            (unsigned long long)(const void*)(img + (size_t)b * CHAN * HW + (size_t)y * WW);
        unsigned int lds_off = (unsigned int)(unsigned long long)(const void*)&smem[0];

        u32x4 g0 = {0u, 0u, 0u, 0u};
        i32x8 g1 = {0, 0, 0, 0, 0, 0, 0, 0};
        i32x4 g2 = {0, 0, 0, 0};
        i32x4 g3 = {0, 0, 0, 0};
        i32x8 g4 = {0, 0, 0, 0, 0, 0, 0, 0};

        g0[0] = 1u;                                       // count = 1 valid descriptor
        g0[1] = lds_off;                                  // lds_addr (bytes)
        g0[2] = (unsigned int)(gaddr & 0xFFFFFFFFull);    // global_addr[31:0]
        g0[3] = (unsigned int)((gaddr >> 32) & 0x01FFFFFFull) | (2u << 30); // addr[56:32], type=2

        g1[0] = (int)(2u << 16);                          // wg_mask=0, data_size=2 (4 bytes)
        g1[1] = (int)(512u << 16);                        // tensor_dim0 = 512  (bits 63:48)
        g1[2] = (int)(3u << 16);                          // tensor_dim1 = 3    (bits 95:80)
        g1[3] = (int)(512u << 16);                        // tile_dim0 = 512    (bits 127:112)
        g1[4] = (int)(3u);                                // tile_dim1 = 3, tile_dim2 = 0
        g1[5] = (int)(HW);                                // tensor_dim0_stride = 262144 elems
        g1[6] = 0;                                        // stride0 hi / stride1 lo
        g1[7] = 0;

        __builtin_amdgcn_tensor_load_to_lds(g0, g1, g2, g3, g4, 0);
        __builtin_amdgcn_s_wait_tensorcnt(0);
    }
    __syncthreads();

    // channel-max row
    for (int i = tid; i < 512; i += 256) {
        float m = fmaxf(smem[i], fmaxf(smem[512 + i], smem[1024 + i]));
        smem[1536 + i] = m;
    }
    __syncthreads();

    const size_t orow = (size_t)bid * 512;
    for (int x = tid; x < 512; x += 256) {
        int lo = x - RAD; if (lo < 0)   lo = 0;
        int hi = x + RAD; if (hi > 511) hi = 511;
        float mx = -FINF, n0 = FINF, n1 = FINF, n2 = FINF;
        for (int k = lo; k <= hi; ++k) {
            mx = fmaxf(mx, smem[1536 + k]);
            n0 = fminf(n0, smem[k]);
            n1 = fminf(n1, smem[512 + k]);
            n2 = fminf(n2, smem[1024 + k]);
        }
        hmax[orow + x] = mx;
        const size_t base = (size_t)b * CHAN * HW + (size_t)y * WW + x;
        hmin[base]          = n0;
        hmin[base + HW]     = n1;
        hmin[base + 2 * HW] = n2;
    }
}

// =====================================================================
// Kernel 2: vertical 35-tap max of hmax -> bc  (bright channel of img)
// =====================================================================
__global__ __launch_bounds__(256) void bcp_vmax(const float* __restrict__ hmax,
                                                float* __restrict__ bc) {
    const int stride = gridDim.x * blockDim.x;
    for (int p = blockIdx.x * blockDim.x + threadIdx.x; p < NPIX; p += stride) {
        const int x = p & 511;
        const int y = (p >> 9) & 511;
        const int b = p >> 18;
        int lo = y - RAD; if (lo < 0)   lo = 0;
        int hi = y + RAD; if (hi > 511) hi = 511;
        const float* col = hmax + (size_t)b * HW + x;
        float mx = -FINF;
        for (int yy = lo; yy <= hi; ++yy) mx = fmaxf(mx, col[(size_t)yy * WW]);
        bc[p] = mx;
    }
}

// =====================================================================
// Kernel 3: per-batch 26 smallest bc values -> pixel indices.
// Keys packed as (float_bits<<32)|index; bc >= 0 so bit order == value
// order, ties resolve to the lowest index (JAX top_k behavior).
// =====================================================================
__global__ __launch_bounds__(256) void bcp_topk(const float* __restrict__ bc,
                                                int* __restrict__ tidx) {
    const int b   = blockIdx.x;
    const int tid = threadIdx.x;

    unsigned long long kb[KSEL];
#pragma unroll
    for (int i = 0; i < KSEL; ++i) kb[i] = ~0ull;

    const float* p = bc + (size_t)b * HW;
    for (int i = tid; i < HW; i += 256) {
        unsigned long long key =
            ((unsigned long long)__float_as_uint(p[i]) << 32) | (unsigned int)i;
        if (key < kb[KSEL - 1]) {
            int j = KSEL - 1;
            while (j > 0 && kb[j - 1] > key) { kb[j] = kb[j - 1]; --j; }
            kb[j] = key;
        }
    }

    __shared__ unsigned long long cand[256 * KSEL];
    __shared__ unsigned long long rv[256];
    __shared__ int ri[256];
    __shared__ int winner;
    for (int i = 0; i < KSEL; ++i) cand[tid * KSEL + i] = kb[i];
    __syncthreads();

    int head = 0;
    for (int it = 0; it < KSEL; ++it) {
        rv[tid] = (head < KSEL) ? cand[tid * KSEL + head] : ~0ull;
        ri[tid] = tid;
        __syncthreads();
        for (int s = 128; s > 0; s >>= 1) {
            if (tid < s && rv[tid + s] < rv[tid]) { rv[tid] = rv[tid + s]; ri[tid] = ri[tid + s]; }
            __syncthreads();
        }
        if (tid == 0) {
            winner = ri[0];
            tidx[b * KSEL + it] = (int)(rv[0] & 0xFFFFFFFFull);
        }
        __syncthreads();
        if (tid == winner) ++head;
        __syncthreads();
    }
}

// =====================================================================
// Kernel 4: airlight. Reference gathers ALL batches' indices (416) for
// each batch: A[b,c] = mean over the union; store 1/(1-A+1e-6).
// =====================================================================
__global__ __launch_bounds__(256) void bcp_airlight(const float* __restrict__ img,
                                                    const int* __restrict__ tidx,
                                                    float* __restrict__ invs) {
    const int b   = blockIdx.x;
    const int tid = threadIdx.x;
    __shared__ float red[3][256];
    float s0 = 0.f, s1 = 0.f, s2 = 0.f;
    const float* base = img + (size_t)b * CHAN * HW;
    for (int j = tid; j < NIDX; j += 256) {
        const int idx = tidx[j];
        s0 += base[idx];
        s1 += base[HW + idx];
        s2 += base[2 * HW + idx];
    }
    red[0][tid] = s0; red[1][tid] = s1; red[2][tid] = s2;
    __syncthreads();
    for (int st = 128; st > 0; st >>= 1) {
        if (tid < st) {
            red[0][tid] += red[0][tid + st];
            red[1][tid] += red[1][tid + st];
            red[2][tid] += red[2][tid + st];
        }
        __syncthreads();
    }
    if (tid < 3) {
        const float A = red[tid][0] * (1.0f / (float)NIDX);
        invs[b * 3 + tid] = 1.0f / (1.0f - A + 1e-6f);
    }
}

// =====================================================================
// Kernel 5: vertical 35-tap per-channel min + t_slide + smooth-L1 loss.
// Block partial sums reduced with V_WMMA_F32_16X16X4_F32 (ones-matrix
// column sum, exact f32), then 8x16 colsums folded in LDS.
// =====================================================================
__global__ __launch_bounds__(256) void bcp_loss(const float* __restrict__ T,
                                                const float* __restrict__ hmin,
                                                const float* __restrict__ invs,
                                                float* __restrict__ partial) {
    float acc = 0.0f;
    const int stride = gridDim.x * blockDim.x;
    for (int p = blockIdx.x * blockDim.x + threadIdx.x; p < NPIX; p += stride) {
        const int x = p & 511;
        const int y = (p >> 9) & 511;
        const int b = p >> 18;
        int lo = y - RAD; if (lo < 0)   lo = 0;
        int hi = y + RAD; if (hi > 511) hi = 511;
        const float* c0 = hmin + (size_t)b * CHAN * HW + x;
        float m0 = FINF, m1 = FINF, m2 = FINF;
        for (int yy = lo; yy <= hi; ++yy) {
            const size_t o = (size_t)yy * WW;
            m0 = fminf(m0, c0[o]);
            m1 = fminf(m1, c0[o + HW]);
            m2 = fminf(m2, c0[o + 2 * HW]);
        }
        const float s0 = invs[b * 3 + 0];
        const float s1 = invs[b * 3 + 1];
        const float s2 = invs[b * 3 + 2];
        const float bc2 = fmaxf((1.0f - m0) * s0,
                          fmaxf((1.0f - m1) * s1, (1.0f - m2) * s2));
        const float t  = 1.0f - 0.95f * bc2;
        const float d  = T[p] - t;
        const float ad = fabsf(d);
        acc += (ad < 1.0f) ? 0.5f * d * d : (ad - 0.5f);

        const int pn = p + stride;                 // prefetch next T element
        if (pn < NPIX) __builtin_prefetch(T + pn, 0, 1);
    }

    // --- wave-level exact f32 sum via WMMA with ones A-matrix ---
    __shared__ float wred[128];
    f32x2 av = {1.0f, 1.0f};          // A = ones(16x4)
    f32x2 bv = {acc, 0.0f};           // B holds one partial per lane
    f32x8 cv = {};
    cv = __builtin_amdgcn_wmma_f32_16x16x4_f32(false, av, false, bv,
                                               (short)0, cv, false, false);
    const int lane = threadIdx.x & 31;
    const int wv   = threadIdx.x >> 5;
    if (lane < 16) wred[wv * 16 + lane] = cv[0];   // row M=0: colsum[n] in lane n
    __syncthreads();
    if (threadIdx.x == 0) {
        float s = 0.0f;
        for (int i = 0; i < 128; ++i) s += wred[i];
        partial[blockIdx.x] = s;
    }
}

// =====================================================================
// Kernel 6: final reduction -> loss scalar
// =====================================================================
__global__ __launch_bounds__(256) void bcp_finish(const float* __restrict__ partial,
                                                  int n, float* __restrict__ out) {
    __shared__ float red[256];
    float s = 0.0f;
    for (int i = threadIdx.x; i < n; i += 256) s += partial[i];
    red[threadIdx.x] = s;
    __syncthreads();
    for (int st = 128; st > 0; st >>= 1) {
        if (threadIdx.x < st) red[threadIdx.x] += red[threadIdx.x + st];
        __syncthreads();
    }
    if (threadIdx.x == 0) out[0] = red[0] * (1.0f / (float)NPIX);
}

// =====================================================================
extern "C" void kernel_launch(void* const* d_in, const int* in_sizes, int n_in,
                              void* d_out, int out_size, void* d_ws, size_t ws_size,
                              hipStream_t stream) {
    const float* img = (const float*)d_in[0];   // (16,3,512,512) f32
    const float* T   = (const float*)d_in[1];   // (16,1,512,512) f32
    float* out = (float*)d_out;                 // scalar f32

    float* ws      = (float*)d_ws;
    float* hmax    = ws;                                   // 16*HW floats
    float* hmin    = hmax + (size_t)BATCH * HW;            // 48*HW floats
    float* bc      = hmin + (size_t)BATCH * CHAN * HW;     // 16*HW floats
    float* invs    = bc + (size_t)BATCH * HW;              // 48 floats (pad to 64)
    int*   tidx    = (int*)(invs + 64);                    // 416 ints (pad to 512)
    float* partial = (float*)(tidx + 512);                 // 4096 floats

    bcp_hpass   <<<BATCH * HH, 256, 0, stream>>>(img, hmax, hmin);
    bcp_vmax    <<<4096, 256, 0, stream>>>(hmax, bc);
    bcp_topk    <<<BATCH, 256, 0, stream>>>(bc, tidx);
    bcp_airlight<<<BATCH, 256, 0, stream>>>(img, tidx, invs);
    bcp_loss    <<<4096, 256, 0, stream>>>(T, hmin, invs, partial);
    bcp_finish  <<<1, 256, 0, stream>>>(partial, 4096, out);
}